// SkeletonizeDiff_76862734729381
// MI455X (gfx1250) — compile-verified
//
#include <hip/hip_runtime.h>
#include <cstdint>
#include <cstddef>

// Soft-skeletonization (5 iters) over [4,128,256,256] fp32.
// Fused per-iteration kernel: a_{k+1} = erode(a_k) and skel-update with
// dilate(a_{k+1}), computed entirely in LDS from one async-DMA'd input tile.
// CDNA5 path: global_load_async_to_lds_b128 (ASYNCcnt-tracked 16B DMA),
// s_wait_asynccnt, wave32 LDS tiling, non-temporal skel traffic.

#define W_DIM 256
#define H_DIM 256
#define D_DIM 128
#define B_DIM 4

constexpr int TX = 32, TY = 8, TZ = 8;                   // output tile
// x-halo widened to 4 so every 4-float group is 16B-aligned in global memory.
constexpr int PX = TX + 8, PY = TY + 4, PZ = TZ + 4;     // p tile: 40x12x12
constexpr int MXX = TX + 2, MXY = TY + 4, MXZ = TZ + 4;  // after x-min: 34x12x12
constexpr int EX = TX + 2, EY = TY + 2, EZ = TZ + 2;     // eroded + halo1: 34x10x10
constexpr int DXX = TX, DXY = TY + 2, DXZ = TZ + 2;      // after x-max: 32x10x10

constexpr int PN  = PX * PY * PZ;     // 5760 (row stride 160B, 16B-aligned)
constexpr int MXN = MXX * MXY * MXZ;  // 4896
constexpr int EN  = EX * EY * EZ;     // 3400
constexpr int DXN = DXX * DXY * DXZ;  // 3200
constexpr int ON  = TX * TY * TZ;     // 2048
constexpr int NTHREADS = 256;         // 8 wave32

__global__ __launch_bounds__(NTHREADS)
void skel_step(const float* __restrict__ in, float* __restrict__ skel,
               float* __restrict__ a_out, int init)
{
  __shared__ __align__(16) float sp[PN];  // input tile (x-halo 4, y/z-halo 2)
  __shared__ float smx[MXN];              // x-min pass; reused as x-max pass
  __shared__ float ser[EN];               // eroded tile (halo 1)

  const int tid = threadIdx.x;
  const int x0 = blockIdx.x * TX;
  const int y0 = blockIdx.y * TY;
  const int zblk = blockIdx.z % (D_DIM / TZ);
  const int b = blockIdx.z / (D_DIM / TZ);
  const int z0 = zblk * TZ;
  const size_t vbase = (size_t)b * D_DIM * H_DIM * W_DIM;
  const float* inb = in + vbase;

  const unsigned sp_lds = (unsigned)(size_t)(&sp[0]);

  // ---- Stage 0: async DMA p-tile into LDS, 16B per op; OOB -> +inf --------
  // Groups of 4 contiguous x-elements start at gx % 4 == 0; W % 4 == 0, so a
  // group is either fully inside [0,W) or fully outside -> exact OOB test.
  for (int g = tid; g < PN / 4; g += NTHREADS) {
    int i = g * 4;
    int lz = i / (PY * PX);
    int r = i - lz * (PY * PX);
    int ly = r / PX;
    int lx = r - ly * PX;  // multiple of 4
    int gx = x0 - 4 + lx, gy = y0 - 2 + ly, gz = z0 - 2 + lz;
    if ((unsigned)gx < (unsigned)W_DIM && (unsigned)gy < (unsigned)H_DIM &&
        (unsigned)gz < (unsigned)D_DIM) {
      int boff = ((gz * H_DIM + gy) * W_DIM + gx) * 4;  // 16B-aligned
      unsigned lds = sp_lds + (unsigned)i * 4u;         // 16B-aligned
      // CDNA5 async DMA: copies 16 bytes from global inb + boff to the LDS address in lds; ASYNCcnt
      asm volatile("global_load_async_to_lds_b128 %0, %1, %2 offset:0"
                   :: "v"(lds), "v"(boff), "s"(inb)
                   : "memory");
    } else {
      const float inf = __builtin_inff();  // identity for min over valid window
      float4 f = make_float4(inf, inf, inf, inf);
      *(float4*)&sp[i] = f;
    }
  }
  asm volatile("s_wait_asynccnt 0x0" ::: "memory");
  __syncthreads();

  // ---- Stage 1: separable x-min (center gx = x0-1+lx reads sp lx+2..lx+4) -
  for (int i = tid; i < MXN; i += NTHREADS) {
    int lz = i / (MXY * MXX);
    int r = i - lz * (MXY * MXX);
    int ly = r / MXX;
    int lx = r - ly * MXX;
    const float* rp = &sp[(lz * PY + ly) * PX + lx + 2];
    smx[i] = fminf(fminf(rp[0], rp[1]), rp[2]);
  }
  __syncthreads();

  // ---- Stage 2: 3x3 (y,z) min -> eroded; OOB centers -> -inf --------------
  for (int i = tid; i < EN; i += NTHREADS) {
    int lz = i / (EY * EX);
    int r = i - lz * (EY * EX);
    int ly = r / EX;
    int lx = r - ly * EX;
    int gx = x0 - 1 + lx, gy = y0 - 1 + ly, gz = z0 - 1 + lz;
    float v;
    if ((unsigned)gx < (unsigned)W_DIM && (unsigned)gy < (unsigned)H_DIM &&
        (unsigned)gz < (unsigned)D_DIM) {
      const float* c0 = &smx[(lz * MXY + ly) * MXX + lx];
      const float* c1 = c0 + MXY * MXX;
      const float* c2 = c1 + MXY * MXX;
      float m0 = fminf(fminf(c0[0], c0[MXX]), c0[2 * MXX]);
      float m1 = fminf(fminf(c1[0], c1[MXX]), c1[2 * MXX]);
      float m2 = fminf(fminf(c2[0], c2[MXX]), c2[2 * MXX]);
      v = fminf(fminf(m0, m1), m2);
    } else {
      v = -__builtin_inff();  // identity for max over valid window
    }
    ser[i] = v;
  }
  __syncthreads();

  // ---- Stage 3: separable x-max of eroded (reuse smx LDS) -----------------
  float* sdx = smx;
  for (int i = tid; i < DXN; i += NTHREADS) {
    int lz = i / (DXY * DXX);
    int r = i - lz * (DXY * DXX);
    int ly = r / DXX;
    int lx = r - ly * DXX;
    const float* rp = &ser[(lz * EY + ly) * EX + lx];
    sdx[i] = fmaxf(fmaxf(rp[0], rp[1]), rp[2]);
  }
  __syncthreads();

  // ---- Stage 4: 3x3 (y,z) max -> open; fused skel update + a_{k+1} store --
  for (int i = tid; i < ON; i += NTHREADS) {
    int lz = i / (TY * TX);
    int r = i - lz * (TY * TX);
    int ly = r / TX;
    int lx = r - ly * TX;
    const float* c0 = &sdx[(lz * DXY + ly) * DXX + lx];
    const float* c1 = c0 + DXY * DXX;
    const float* c2 = c1 + DXY * DXX;
    float o = fmaxf(fmaxf(c0[0], c0[DXX]), c0[2 * DXX]);
    o = fmaxf(o, fmaxf(fmaxf(c1[0], c1[DXX]), c1[2 * DXX]));
    o = fmaxf(o, fmaxf(fmaxf(c2[0], c2[DXX]), c2[2 * DXX]));

    float pc = sp[((lz + 2) * PY + (ly + 2)) * PX + (lx + 4)];   // a_k center
    float ec = ser[((lz + 1) * EY + (ly + 1)) * EX + (lx + 1)];  // a_{k+1} center

    size_t gidx = vbase +
        ((size_t)(z0 + lz) * H_DIM + (size_t)(y0 + ly)) * W_DIM + (size_t)(x0 + lx);
    a_out[gidx] = ec;  // regular store: stays hot in 192MB L2 for next launch

    float d = fmaxf(pc - o, 0.0f);  // relu(a_k - dilate(erode(a_k)))
    if (init) {
      __builtin_nontemporal_store(d, &skel[gidx]);
    } else {
      float s = __builtin_nontemporal_load(&skel[gidx]);
      __builtin_nontemporal_store(s + fmaxf(d - s * d, 0.0f), &skel[gidx]);
    }
  }
}

extern "C" void kernel_launch(void* const* d_in, const int* in_sizes, int n_in,
                              void* d_out, int out_size, void* d_ws, size_t ws_size,
                              hipStream_t stream)
{
  (void)in_sizes; (void)n_in; (void)out_size; (void)ws_size;
  const size_t NVOX = (size_t)B_DIM * D_DIM * H_DIM * W_DIM;

  const float* p = (const float*)d_in[0];
  float* skel = (float*)d_out;
  float* bufA = (float*)d_ws;          // a-volume ping
  float* bufB = bufA + NVOX;           // a-volume pong (needs 2*128MiB of ws)

  dim3 grid(W_DIM / TX, H_DIM / TY, (D_DIM / TZ) * B_DIM);
  dim3 block(NTHREADS);

  // k=0: skel = relu(p - open(p)), a1 = erode(p)
  // k=1..5: a_{k+1} = erode(a_k); delta = relu(a_k - dilate(a_{k+1}));
  //         skel += relu(delta - skel*delta)
  const float* cur = p;
  float* nxt = bufA;
  for (int k = 0; k < 6; ++k) {
    skel_step<<<grid, block, 0, stream>>>(cur, skel, nxt, (k == 0) ? 1 : 0);
    cur = nxt;
    nxt = (nxt == bufA) ? bufB : bufA;
  }
}